// XMILoss_33835752358351
// MI455X (gfx1250) — compile-verified
//
#include <hip/hip_runtime.h>
#include <hip/hip_bf16.h>
#include <math.h>

typedef __attribute__((ext_vector_type(2))) float v2f;
typedef __attribute__((ext_vector_type(8))) float v8f;

#define BATCH   16384
#define NCLS    1000
#define CFEAT   1280

// ---------------- reduction helpers (wave32) ----------------
__device__ __forceinline__ float waveReduceSum(float v) {
#pragma unroll
    for (int o = 16; o > 0; o >>= 1) v += __shfl_down(v, o, 32);
    return v;
}
__device__ __forceinline__ float waveReduceMax(float v) {
#pragma unroll
    for (int o = 16; o > 0; o >>= 1) v = fmaxf(v, __shfl_down(v, o, 32));
    return v;
}
__device__ __forceinline__ float blockReduceSum(float v, float* s) {
    int lane = threadIdx.x & 31, w = threadIdx.x >> 5;
    v = waveReduceSum(v);
    __syncthreads();
    if (lane == 0) s[w] = v;
    __syncthreads();
    int nw = (blockDim.x + 31) >> 5;
    v = (threadIdx.x < (unsigned)nw) ? s[threadIdx.x] : 0.0f;
    if (w == 0) v = waveReduceSum(v);
    return v;  // valid in thread 0
}
__device__ __forceinline__ float blockReduceMax(float v, float* s) {
    int lane = threadIdx.x & 31, w = threadIdx.x >> 5;
    v = waveReduceMax(v);
    __syncthreads();
    if (lane == 0) s[w] = v;
    __syncthreads();
    int nw = (blockDim.x + 31) >> 5;
    v = (threadIdx.x < (unsigned)nw) ? s[threadIdx.x] : -INFINITY;
    if (w == 0) v = waveReduceMax(v);
    return v;  // valid in thread 0
}

// ---------------- gfx1250 async global->LDS copy (ASYNCcnt-tracked) ----------------
__device__ __forceinline__ void async_copy_b128(unsigned lds_off, const void* gaddr) {
    unsigned long long ga = (unsigned long long)gaddr;
    asm volatile("global_load_async_to_lds_b128 %0, %1, off"
                 :: "v"(lds_off), "v"(ga)
                 : "memory");
}

// ---------------- zero accumulators + mu ----------------
__global__ void zero_kernel(float* p, int n) {
    for (int i = blockIdx.x * blockDim.x + threadIdx.x; i < n; i += gridDim.x * blockDim.x)
        p[i] = 0.0f;
}

// ---------------- column sums of features (for mean) ----------------
__global__ void colsum_kernel(const float* __restrict__ F, float* __restrict__ muSum) {
    int c = blockIdx.x * 256 + threadIdx.x;          // 0..1279
    int b0 = blockIdx.y * (BATCH / 64);
    float s = 0.0f;
    for (int b = b0; b < b0 + (BATCH / 64); ++b)
        s += F[(size_t)b * CFEAT + c];
    atomicAdd(muSum + c, s);
}

// ---------------- NLL gather ----------------
__global__ void nll_kernel(const float* __restrict__ out, const int* __restrict__ tgt,
                           float* __restrict__ ws) {
    __shared__ float red[32];
    float local = 0.0f;
    for (int b = blockIdx.x * blockDim.x + threadIdx.x; b < BATCH; b += gridDim.x * blockDim.x)
        local += out[(size_t)b * NCLS + tgt[b]];
    local = blockReduceSum(local, red);
    if (threadIdx.x == 0) atomicAdd(ws + 0, local);
}

// ---------------- KL term: per-row log-softmax stats ----------------
__global__ void kl_kernel(const float* __restrict__ F, float* __restrict__ ws) {
    __shared__ float row[CFEAT];
    __shared__ float red[32];
    __shared__ float sm;
    const float* f = F + (size_t)blockIdx.x * CFEAT;
    float lmax = -INFINITY, sF = 0.0f;
    for (int c = threadIdx.x; c < CFEAT; c += blockDim.x) {
        float v = f[c];
        row[c] = v;
        lmax = fmaxf(lmax, v);
        sF += v;
    }
    __syncthreads();
    float m = blockReduceMax(lmax, red);
    if (threadIdx.x == 0) sm = m;
    __syncthreads();
    m = sm;
    float sE = 0.0f, sEX = 0.0f;
    for (int c = threadIdx.x; c < CFEAT; c += blockDim.x) {
        float d = row[c] - m;
        float e = __expf(d);
        sE += e;
        sEX += e * d;
    }
    sF  = blockReduceSum(sF, red);
    sE  = blockReduceSum(sE, red);
    sEX = blockReduceSum(sEX, red);
    if (threadIdx.x == 0) {
        float logS = __logf(sE);
        float sumL  = sF - (float)CFEAT * (m + logS);   // sum_c L[b,c]
        float sumPL = sEX / sE - logS;                  // sum_c P*L
        atomicAdd(ws + 1, sumPL);
        atomicAdd(ws + 2, sumL);
    }
}

// ---------------- cov GEMM: G = F^T F via fp32 WMMA + async-LDS double buffering ----------
// Block macro-tile: 64 (i) x 128 (j) of G; 8 waves arranged 2x4, each 32x32.
// K-slab = 32 rows of F staged to LDS per buffer via global_load_async_to_lds_b128.
// Fused epilogue: ||G/B - mu mu^T - I||_F^2 partial -> atomicAdd.
#define KSLAB   32
#define A_STR   72      // 64 + 8 pad floats (bank-conflict-free half-wave reads)
#define B_STR   136     // 128 + 8 pad floats

__global__ __launch_bounds__(256) void cov_gemm_kernel(const float* __restrict__ F,
                                                       const float* __restrict__ muSum,
                                                       float* __restrict__ ws) {
    __shared__ float shA[2][KSLAB * A_STR];   // 2 x 9216 B
    __shared__ float shB[2][KSLAB * B_STR];   // 2 x 17408 B

    const int t    = threadIdx.x;
    const int wave = t >> 5;
    const int lane = t & 31;
    const int half = lane >> 4, l16 = lane & 15;

    const int bi = blockIdx.x / 10;           // 0..19  -> i0 = bi*64
    const int bj = blockIdx.x % 10;           // 0..9   -> j0 = bj*128
    const int i0 = bi * 64, j0 = bj * 128;
    const int ia = (wave >> 2) * 32;          // wave i-offset within A panel
    const int jb = (wave & 3) * 32;           // wave j-offset within B panel

    const unsigned offA[2] = { (unsigned)(size_t)&shA[0][0], (unsigned)(size_t)&shA[1][0] };
    const unsigned offB[2] = { (unsigned)(size_t)&shB[0][0], (unsigned)(size_t)&shB[1][0] };

    // stage one K-slab (32 rows) into buffer `buf`: 6 async b128 issues per thread
    auto stage = [&](int buf, int k) {
        // A panel: rows k..k+31, cols i0..i0+63 (256 B/row, 16 thr/row, 16 rows/instr)
#pragma unroll
        for (int n = 0; n < 2; ++n) {
            int row  = 16 * n + (t >> 4);
            int colB = (t & 15) * 16;
            unsigned lo = offA[buf] + (unsigned)(row * (A_STR * 4) + colB);
            const char* g = (const char*)(F + (size_t)(k + row) * CFEAT + i0) + colB;
            async_copy_b128(lo, g);
        }
        // B panel: rows k..k+31, cols j0..j0+127 (512 B/row, 32 thr/row, 8 rows/instr)
#pragma unroll
        for (int n = 0; n < 4; ++n) {
            int row  = 8 * n + (t >> 5);
            int colB = (t & 31) * 16;
            unsigned lo = offB[buf] + (unsigned)(row * (B_STR * 4) + colB);
            const char* g = (const char*)(F + (size_t)(k + row) * CFEAT + j0) + colB;
            async_copy_b128(lo, g);
        }
    };

    v8f acc00 = {}, acc01 = {}, acc10 = {}, acc11 = {};

    const int NS = BATCH / KSLAB;             // 512 slabs
    stage(0, 0);
    for (int s = 0; s < NS; ++s) {
        int cur = s & 1;
        if (s + 1 < NS) {
            stage(1 - cur, (s + 1) * KSLAB);
            asm volatile("s_wait_asynccnt 6" ::: "memory");   // slab s done, s+1 in flight
        } else {
            asm volatile("s_wait_asynccnt 0" ::: "memory");
        }
        __syncthreads();

        const float* Ab = shA[cur];
        const float* Bb = shB[cur];
#pragma unroll
        for (int kk = 0; kk < KSLAB; kk += 4) {
            int kr = kk + 2 * half;
            const float* ar0 = Ab + kr * A_STR;
            const float* ar1 = ar0 + A_STR;
            const float* br0 = Bb + kr * B_STR;
            const float* br1 = br0 + B_STR;
            v2f a0, a1, b0, b1;
            a0.x = ar0[ia + l16];      a0.y = ar1[ia + l16];
            a1.x = ar0[ia + 16 + l16]; a1.y = ar1[ia + 16 + l16];
            b0.x = br0[jb + l16];      b0.y = br1[jb + l16];
            b1.x = br0[jb + 16 + l16]; b1.y = br1[jb + 16 + l16];
            acc00 = __builtin_amdgcn_wmma_f32_16x16x4_f32(false, a0, false, b0, (short)0, acc00, false, false);
            acc01 = __builtin_amdgcn_wmma_f32_16x16x4_f32(false, a0, false, b1, (short)0, acc01, false, false);
            acc10 = __builtin_amdgcn_wmma_f32_16x16x4_f32(false, a1, false, b0, (short)0, acc10, false, false);
            acc11 = __builtin_amdgcn_wmma_f32_16x16x4_f32(false, a1, false, b1, (short)0, acc11, false, false);
        }
        __syncthreads();                       // buffer `cur` free for restage
    }

    // fused epilogue: cov = G/B - mu mu^T ; accumulate ||cov - I||^2
    const float invB = 1.0f / (float)BATCH;
    const int gi0 = i0 + ia, gj0 = j0 + jb;
    int jA = gj0 + l16, jB = gj0 + 16 + l16;
    float mjA = muSum[jA] * invB;
    float mjB = muSum[jB] * invB;
    float local = 0.0f;
#pragma unroll
    for (int r = 0; r < 8; ++r) {
        int iA = gi0 + r + 8 * half;
        int iB = iA + 16;
        float miA = muSum[iA] * invB;
        float miB = muSum[iB] * invB;
        float v;
        v = acc00[r] * invB - miA * mjA - (iA == jA ? 1.0f : 0.0f); local += v * v;
        v = acc01[r] * invB - miA * mjB - (iA == jB ? 1.0f : 0.0f); local += v * v;
        v = acc10[r] * invB - miB * mjA - (iB == jA ? 1.0f : 0.0f); local += v * v;
        v = acc11[r] * invB - miB * mjB - (iB == jB ? 1.0f : 0.0f); local += v * v;
    }
    local = waveReduceSum(local);
    if (lane == 0) atomicAdd(ws + 3, local);
}

// ---------------- ortho GEMM: H = W W^T via fp32 WMMA, fused ||H - I||^2 ----------------
__global__ __launch_bounds__(256) void ortho_gemm_kernel(const float* __restrict__ W,
                                                         float* __restrict__ ws) {
    int wave = threadIdx.x >> 5;
    int lane = threadIdx.x & 31;
    int tile = blockIdx.x * 8 + wave;           // 0..1023
    int ti = tile >> 5, tj = tile & 31;
    int i0 = ti * 32, j0 = tj * 32;
    int half = lane >> 4, l16 = lane & 15;

    int rA = i0 + l16,      rB = i0 + 16 + l16;   // A rows (rows of H)
    int cA = j0 + l16,      cB = j0 + 16 + l16;   // B rows (cols of H)
    bool okA = rA < NCLS, okB = rB < NCLS, okC = cA < NCLS, okD = cB < NCLS;
    const float* pA = W + (size_t)(okA ? rA : NCLS - 1) * CFEAT;
    const float* pB = W + (size_t)(okB ? rB : NCLS - 1) * CFEAT;
    const float* pC = W + (size_t)(okC ? cA : NCLS - 1) * CFEAT;
    const float* pD = W + (size_t)(okD ? cB : NCLS - 1) * CFEAT;
    const v2f vz = {0.0f, 0.0f};

    v8f acc00 = {}, acc01 = {}, acc10 = {}, acc11 = {};
    for (int k = 0; k < CFEAT; k += 4) {
        int kk = k + 2 * half;                  // contiguous 8B pair along K
        v2f a0 = *(const v2f*)(pA + kk); if (!okA) a0 = vz;
        v2f a1 = *(const v2f*)(pB + kk); if (!okB) a1 = vz;
        v2f b0 = *(const v2f*)(pC + kk); if (!okC) b0 = vz;
        v2f b1 = *(const v2f*)(pD + kk); if (!okD) b1 = vz;
        acc00 = __builtin_amdgcn_wmma_f32_16x16x4_f32(false, a0, false, b0, (short)0, acc00, false, false);
        acc01 = __builtin_amdgcn_wmma_f32_16x16x4_f32(false, a0, false, b1, (short)0, acc01, false, false);
        acc10 = __builtin_amdgcn_wmma_f32_16x16x4_f32(false, a1, false, b0, (short)0, acc10, false, false);
        acc11 = __builtin_amdgcn_wmma_f32_16x16x4_f32(false, a1, false, b1, (short)0, acc11, false, false);
    }

    int jA = j0 + l16, jB = j0 + 16 + l16;
    float local = 0.0f;
#pragma unroll
    for (int r = 0; r < 8; ++r) {
        int iA = i0 + r + 8 * half;
        int iB = iA + 16;
        float v;
        if (iA < NCLS && jA < NCLS) { v = acc00[r] - (iA == jA ? 1.0f : 0.0f); local += v * v; }
        if (iA < NCLS && jB < NCLS) { v = acc01[r] - (iA == jB ? 1.0f : 0.0f); local += v * v; }
        if (iB < NCLS && jA < NCLS) { v = acc10[r] - (iB == jA ? 1.0f : 0.0f); local += v * v; }
        if (iB < NCLS && jB < NCLS) { v = acc11[r] - (iB == jB ? 1.0f : 0.0f); local += v * v; }
    }
    local = waveReduceSum(local);
    if (lane == 0) atomicAdd(ws + 4, local);
}

// ---------------- L1 / L2 over weights ----------------
__global__ void l1l2_kernel(const float* __restrict__ W, float* __restrict__ ws) {
    __shared__ float red[32];
    float s1 = 0.0f, s2 = 0.0f;
    const int n = NCLS * CFEAT;
    for (int i = blockIdx.x * blockDim.x + threadIdx.x; i < n; i += gridDim.x * blockDim.x) {
        float w = W[i];
        s1 += fabsf(w);
        s2 += w * w;
    }
    s1 = blockReduceSum(s1, red);
    s2 = blockReduceSum(s2, red);
    if (threadIdx.x == 0) { atomicAdd(ws + 5, s1); atomicAdd(ws + 6, s2); }
}

// ---------------- combine ----------------
__global__ void finalize_kernel(const float* __restrict__ ws, float* __restrict__ out) {
    float nll   = -ws[0] / (float)BATCH;
    float kl    = ((float)CFEAT * ws[1] - ws[2]) / (float)BATCH;
    float cov   = ws[3];
    float ortho = ws[4];
    float l1    = ws[5];
    float l2    = sqrtf(ws[6]);
    out[0] = 1.0f * nll + 0.2f * kl + 0.2f * cov + 0.1f * ortho + 0.1f * l1 + 0.1f * l2;
}

extern "C" void kernel_launch(void* const* d_in, const int* in_sizes, int n_in,
                              void* d_out, int out_size, void* d_ws, size_t ws_size,
                              hipStream_t stream) {
    const float* output   = (const float*)d_in[0];   // [16384, 1000] log-probs
    const int*   target   = (const int*)  d_in[1];   // [16384]
    const float* weights  = (const float*)d_in[2];   // [1000, 1280]
    const float* features = (const float*)d_in[3];   // [16384, 1280]
    float* out = (float*)d_out;

    float* ws = (float*)d_ws;        // [0..6] scalar accumulators
    float* mu = ws + 16;             // [1280] column sums of features

    zero_kernel<<<8, 256, 0, stream>>>(ws, 16 + CFEAT);

    colsum_kernel<<<dim3(5, 64), 256, 0, stream>>>(features, mu);
    nll_kernel<<<64, 256, 0, stream>>>(output, target, ws);
    kl_kernel<<<BATCH, 256, 0, stream>>>(features, ws);
    cov_gemm_kernel<<<200, 256, 0, stream>>>(features, mu, ws);   // 20 x 10 macro-tiles
    ortho_gemm_kernel<<<128, 256, 0, stream>>>(weights, ws);
    l1l2_kernel<<<80, 256, 0, stream>>>(weights, ws);
    finalize_kernel<<<1, 1, 0, stream>>>(ws, out);
}